// RelativePositionAttention_60550448939273
// MI455X (gfx1250) — compile-verified
//
#include <hip/hip_runtime.h>

typedef __attribute__((ext_vector_type(2))) float v2f;
typedef __attribute__((ext_vector_type(8))) float v8f;

#define B_   16
#define N1_  577
#define NKV  576
#define C_   768
#define H_   12
#define HD   64

static __device__ __forceinline__ v8f wmma_f32(v2f a, v2f b, v8f c) {
  // D = A(16x4) x B(4x16) + C, fp32, wave32
  return __builtin_amdgcn_wmma_f32_16x16x4_f32(false, a, false, b, (short)0, c,
                                               false, false);
}

// ---------------------------------------------------------------------------
// Expand bias_table[rel_index] -> bias_full[H][577][577] (zero CLS row/col).
// 16 MB, L2-resident; amortizes the gather across the whole batch (16x).
// ---------------------------------------------------------------------------
__global__ void bias_expand_kernel(const float* __restrict__ table,
                                   const int* __restrict__ rel,
                                   float* __restrict__ bf) {
  int idx = blockIdx.x * blockDim.x + threadIdx.x;
  const int total = H_ * N1_ * N1_;
  if (idx >= total) return;
  int h = idx / (N1_ * N1_);
  int rem = idx - h * (N1_ * N1_);
  int t = rem / N1_;
  int j = rem - t * N1_;
  float v = 0.0f;
  if (t > 0 && j > 0) {
    int ri = rel[(t - 1) * NKV + (j - 1)];
    v = table[ri * H_ + h];
  }
  bf[idx] = v;
}

// ---------------------------------------------------------------------------
// WMMA fp32 GEMM: out(M,N) = A(M,K) @ W(K,N) + bias.
//  - register-prefetch pipeline: next tile's global loads issued before the
//    WMMA chunk loop, waited on only at the next LDS fill.
//  - W tile stored TRANSPOSED in LDS so every A/B fragment is one ds_load_b64
//    (padded stride 18 -> conflict-free 16-lane access).
// mode 0: scatter into q/k/v [B,H,N1,64] (q scaled by 1/8).
// mode 1: plain row-major store to out.
// ---------------------------------------------------------------------------
#define BM 64
#define BN 64
#define BK 16

__global__ __launch_bounds__(256) void gemm_wmma_kernel(
    const float* __restrict__ A, const float* __restrict__ W,
    const float* __restrict__ bias, int M, int N, int K, int mode,
    float* __restrict__ out, float* __restrict__ qo, float* __restrict__ ko,
    float* __restrict__ vo) {
  __shared__ float As[BM][18];   // [m][k], padded
  __shared__ float Wt[BN][18];   // [n][k], transposed + padded

  const int tid = threadIdx.x;
  const int wid = tid >> 5;
  const int lane = tid & 31;
  const int lane15 = lane & 15;
  const int hi = lane >> 4;
  const int k0 = hi * 2;

  const int m0 = blockIdx.y * BM;
  const int n0 = blockIdx.x * BN;
  const int wm = (wid & 3) * 16;   // 4 waves along M
  const int wn = (wid >> 2) * 32;  // 2 waves along N, 2 subtiles each

  v8f acc[2] = {};

  // cooperative tile-load indices (float4 per thread)
  const int ar = tid >> 2;            // 0..63
  const int ac = (tid & 3) * 4;       // 0,4,8,12
  const int arow = min(m0 + ar, M - 1);
  const int wr = tid >> 4;            // 0..15 (k within tile)
  const int wc = (tid & 15) * 4;      // 0..60 (n within tile)

  const float* aptr = A + (size_t)arow * K;
  const float* wptr = W + (size_t)wr * N + n0 + wc;

  // prime the pipeline
  float4 avr = *(const float4*)(aptr + ac);
  float4 wvr = *(const float4*)(wptr);

  for (int kb = 0; kb < K; kb += BK) {
    __syncthreads();  // previous iteration's LDS readers done
    As[ar][ac + 0] = avr.x; As[ar][ac + 1] = avr.y;
    As[ar][ac + 2] = avr.z; As[ar][ac + 3] = avr.w;
    Wt[wc + 0][wr] = wvr.x; Wt[wc + 1][wr] = wvr.y;
    Wt[wc + 2][wr] = wvr.z; Wt[wc + 3][wr] = wvr.w;
    __syncthreads();
    if (kb + BK < K) {  // prefetch next tiles; latency hidden by WMMA loop
      avr = *(const float4*)(aptr + kb + BK + ac);
      wvr = *(const float4*)(wptr + (size_t)(kb + BK) * N);
    }
#pragma unroll
    for (int kk = 0; kk < BK; kk += 4) {
      v2f a  = *(const v2f*)&As[wm + lane15][kk + k0];
      v2f b0 = *(const v2f*)&Wt[wn + lane15][kk + k0];
      v2f b1 = *(const v2f*)&Wt[wn + 16 + lane15][kk + k0];
      acc[0] = wmma_f32(a, b0, acc[0]);
      acc[1] = wmma_f32(a, b1, acc[1]);
    }
  }

#pragma unroll
  for (int s = 0; s < 2; ++s) {
    const int gcol = n0 + wn + s * 16 + lane15;
    const float bv = bias[gcol];
#pragma unroll
    for (int r = 0; r < 8; ++r) {
      const int grow = m0 + wm + r + 8 * hi;
      if (grow < M) {
        float val = acc[s][r] + bv;
        if (mode == 0) {
          int which = gcol / C_;            // 0=q 1=k 2=v
          int rem = gcol - which * C_;
          int h = rem >> 6;
          int d = rem & 63;
          int b = grow / N1_;
          int t = grow - b * N1_;
          size_t di = (((size_t)(b * H_ + h)) * N1_ + t) * HD + d;
          if (which == 0)      qo[di] = val * 0.125f;   // * hd^-0.5
          else if (which == 1) ko[di] = val;
          else                 vo[di] = val;
        } else {
          out[(size_t)grow * N + gcol] = val;
        }
      }
    }
  }
}

// ---------------------------------------------------------------------------
// Fused flash attention: one wave per (b,h, 16-query tile). Streams keys in
// tiles of 16, online softmax, no materialized score matrix. P goes through
// 1 KB LDS to convert WMMA D-layout -> A-layout for the PV product.
// ---------------------------------------------------------------------------
__global__ __launch_bounds__(32) void flash_attn_kernel(
    const float* __restrict__ q, const float* __restrict__ k,
    const float* __restrict__ v, const float* __restrict__ bf,
    float* __restrict__ ctx) {
  __shared__ float P[16][18];

  const int lane = threadIdx.x;
  const int lane15 = lane & 15;
  const int hi = lane >> 4;
  const int k0 = hi * 2;

  const int bh = blockIdx.y;
  const int b = bh / H_;
  const int h = bh - b * H_;
  const int q0 = blockIdx.x * 16;

  const float* qp = q + (size_t)bh * N1_ * HD;
  const float* kp = k + (size_t)bh * N1_ * HD;
  const float* vp = v + (size_t)bh * N1_ * HD;
  const float* bp = bf + (size_t)h * N1_ * N1_;

  // Q fragments for all 16 K-steps (hd=64), held in registers (pre-scaled q)
  const int qrow = min(q0 + lane15, N1_ - 1);
  v2f aq[16];
#pragma unroll
  for (int i = 0; i < 16; ++i)
    aq[i] = *(const v2f*)(qp + (size_t)qrow * HD + i * 4 + k0);

  float mrun[8], lrun[8];
  v8f o[4] = {};
#pragma unroll
  for (int r = 0; r < 8; ++r) { mrun[r] = -3.0e38f; lrun[r] = 0.0f; }

  for (int kt = 0; kt < (N1_ + 15) / 16; ++kt) {
    const int key0 = kt * 16;
    const int krow = min(key0 + lane15, N1_ - 1);

    // S = Q @ K^T  (16x16 tile, 16 wmma steps over hd)
    v8f s = {};
#pragma unroll
    for (int i = 0; i < 16; ++i) {
      v2f bk = *(const v2f*)(kp + (size_t)krow * HD + i * 4 + k0);
      s = wmma_f32(aq[i], bk, s);
    }

    const int j = key0 + lane15;
    const bool jvalid = (j < N1_);

    // bias + mask + online softmax (row-wise, 16-lane half reductions)
#pragma unroll
    for (int r = 0; r < 8; ++r) {
      const int t = q0 + r + 8 * hi;
      const int tb = min(t, N1_ - 1);
      float sv = s[r] + bp[(size_t)tb * N1_ + (jvalid ? j : 0)];
      if (!jvalid) sv = -3.0e38f;
      float mx = sv;
      mx = fmaxf(mx, __shfl_xor(mx, 1, 32));
      mx = fmaxf(mx, __shfl_xor(mx, 2, 32));
      mx = fmaxf(mx, __shfl_xor(mx, 4, 32));
      mx = fmaxf(mx, __shfl_xor(mx, 8, 32));
      const float mnew = fmaxf(mrun[r], mx);
      const float fac = __expf(mrun[r] - mnew);
      const float p = __expf(sv - mnew);
      float rs = p;
      rs += __shfl_xor(rs, 1, 32);
      rs += __shfl_xor(rs, 2, 32);
      rs += __shfl_xor(rs, 4, 32);
      rs += __shfl_xor(rs, 8, 32);
      lrun[r] = lrun[r] * fac + rs;
      mrun[r] = mnew;
      o[0][r] = o[0][r] * fac;
      o[1][r] = o[1][r] * fac;
      o[2][r] = o[2][r] * fac;
      o[3][r] = o[3][r] * fac;
      P[r + 8 * hi][lane15] = p;   // D-layout -> row-major staging
    }
    __syncthreads();

    // O += P @ V  (16 wmma: 4 K-chunks x 4 dim-subtiles)
#pragma unroll
    for (int i = 0; i < 4; ++i) {
      v2f ap = *(const v2f*)&P[lane15][i * 4 + k0];
      const int kr0 = min(key0 + i * 4 + k0, N1_ - 1);
      const int kr1 = min(key0 + i * 4 + k0 + 1, N1_ - 1);
#pragma unroll
      for (int sub = 0; sub < 4; ++sub) {
        v2f bv;
        bv.x = vp[(size_t)kr0 * HD + sub * 16 + lane15];
        bv.y = vp[(size_t)kr1 * HD + sub * 16 + lane15];
        o[sub] = wmma_f32(ap, bv, o[sub]);
      }
    }
    __syncthreads();
  }

  // normalize + write ctx[b, t, h*64 + d] (row-major [B,N1,C] for proj GEMM)
#pragma unroll
  for (int r = 0; r < 8; ++r) {
    const int t = q0 + r + 8 * hi;
    if (t < N1_) {
      const float inv = 1.0f / lrun[r];
#pragma unroll
      for (int sub = 0; sub < 4; ++sub) {
        ctx[((size_t)(b * N1_ + t)) * C_ + h * HD + sub * 16 + lane15] =
            o[sub][r] * inv;
      }
    }
  }
}

// ---------------------------------------------------------------------------
extern "C" void kernel_launch(void* const* d_in, const int* in_sizes, int n_in,
                              void* d_out, int out_size, void* d_ws,
                              size_t ws_size, hipStream_t stream) {
  (void)in_sizes; (void)n_in; (void)out_size; (void)ws_size;
  const float* x      = (const float*)d_in[0];
  const float* qkv_w  = (const float*)d_in[1];
  const float* qkv_b  = (const float*)d_in[2];
  const float* proj_w = (const float*)d_in[3];
  const float* proj_b = (const float*)d_in[4];
  const float* table  = (const float*)d_in[5];
  const int*   rel    = (const int*)d_in[6];
  float* out = (float*)d_out;

  const size_t per = (size_t)B_ * H_ * N1_ * HD;  // 7,090,176 floats
  float* qbuf = (float*)d_ws;
  float* kbuf = qbuf + per;
  float* vbuf = kbuf + per;
  float* ctx  = vbuf + per;                        // B*N1*C == per
  float* bf   = ctx + per;                         // H*577*577 floats

  {
    const int total = H_ * N1_ * N1_;
    bias_expand_kernel<<<(total + 255) / 256, 256, 0, stream>>>(table, rel, bf);
  }
  {
    dim3 grid((3 * C_) / BN, (B_ * N1_ + BM - 1) / BM);
    gemm_wmma_kernel<<<grid, 256, 0, stream>>>(x, qkv_w, qkv_b, B_ * N1_,
                                               3 * C_, C_, /*mode=*/0, nullptr,
                                               qbuf, kbuf, vbuf);
  }
  {
    dim3 grid((N1_ + 15) / 16, B_ * H_);
    flash_attn_kernel<<<grid, 32, 0, stream>>>(qbuf, kbuf, vbuf, bf, ctx);
  }
  {
    dim3 grid(C_ / BN, (B_ * N1_ + BM - 1) / BM);
    gemm_wmma_kernel<<<grid, 256, 0, stream>>>(ctx, proj_w, proj_b, B_ * N1_,
                                               C_, C_, /*mode=*/1, out, nullptr,
                                               nullptr, nullptr);
  }
}